// ConnectedFilterLayerBySingleThreshold_62079457296803
// MI455X (gfx1250) — compile-verified
//
#include <hip/hip_runtime.h>

// ---------------- problem constants (from reference) ----------------
#define T_TREES   4
#define N_NODES   262144
#define NP1       262145          // + sentinel slot per tree
#define HW_PIX    1048576         // 1024*1024
#define BETA_F    1000.0f
#define CLAMP_V   12.0f

// ---------------- phase-A tiling ----------------
#define BE        8192            // nodes per block (LDS-resident)
#define BPT       (N_NODES / BE)  // 32 blocks per tree
#define ATHREADS  1024
#define KPT       (BE / ATHREADS) // 8 nodes per thread
#define AITERS    13              // ceil(log2(BE))
#define BITERS    6               // >= ceil(log2(BPT)) + margin

struct __align__(8) Pair { float acc; int ptr; };

// ---------------- CDNA5 Tensor Data Mover: 1-row dword tile -> LDS ----------------
typedef unsigned int v4u __attribute__((ext_vector_type(4)));
typedef int          v8i __attribute__((ext_vector_type(8)));
typedef int          v4i __attribute__((ext_vector_type(4)));

#if defined(__AMDGCN__) && __has_builtin(__builtin_amdgcn_tensor_load_to_lds)
#define HAVE_TDM 1
__device__ __forceinline__ void tdm_load_dwords(unsigned ldsAddr, const void* gptr,
                                                unsigned ndw) {
  unsigned long long ga = (unsigned long long)(size_t)gptr;
  // D# group 0 (128b): count=1 | lds_addr | global_addr[56:0] | type=2
  v4u g0;
  g0[0] = 1u;
  g0[1] = ldsAddr;
  g0[2] = (unsigned)(ga & 0xFFFFFFFFull);
  g0[3] = (unsigned)((ga >> 32) & 0x1FFFFFFull) | (2u << 30);
  // D# group 1 (256b): data_size=4B; tensor_dim0=ndw; tensor_dim1=1;
  // tile_dim0=ndw; tile_dim1/2 unused; dim0_stride=ndw
  v8i g1;
  g1[0] = (int)(2u << 16);                                  // workgroup_mask=0, data_size=2(4B)
  g1[1] = (int)((ndw & 0xFFFFu) << 16);                     // tensor_dim0[15:0] @ bits 63:48
  g1[2] = (int)(((ndw >> 16) & 0xFFFFu) | (1u << 16));      // tensor_dim0[31:16] | tensor_dim1=1
  g1[3] = (int)((ndw & 0xFFFFu) << 16);                     // tile_dim0 @ bits 127:112
  g1[4] = 0;                                                // tile_dim1=0, tile_dim2=0
  g1[5] = (int)ndw;                                         // tensor_dim0_stride[31:0]
  g1[6] = 0;
  g1[7] = 0;
  v4i z4 = {0, 0, 0, 0};
#if __clang_major__ >= 23
  v8i z8 = {0, 0, 0, 0, 0, 0, 0, 0};
  __builtin_amdgcn_tensor_load_to_lds(g0, g1, z4, z4, z8, 0);
#else
  __builtin_amdgcn_tensor_load_to_lds(g0, g1, z4, z4, 0);
#endif
}
#endif

// ---------------- Phase A: c-values + intra-block pointer-doubling in LDS ----------------
__global__ __launch_bounds__(ATHREADS) void phaseA_kernel(
    const float* __restrict__ attr, const float* __restrict__ levels,
    const int* __restrict__ parent, const float* __restrict__ thr,
    Pair* __restrict__ outPairs) {
  __shared__ float s_lvl[BE];     // 32 KB
  __shared__ Pair  s_pair[BE];    // 64 KB

  const int b    = blockIdx.x;
  const int t    = blockIdx.y;
  const int base = b * BE;
  const size_t toff = (size_t)t * N_NODES;
  const float thv = thr[0];

#ifdef HAVE_TDM
  if (threadIdx.x == 0) {
    unsigned lds = (unsigned)(size_t)(void*)s_lvl;   // local offset (flat addr truncates)
    tdm_load_dwords(lds, levels + toff + base, BE);  // async DMA: levels block -> LDS
    __builtin_amdgcn_s_wait_tensorcnt(0);
  }
#else
  for (int k = 0; k < KPT; ++k) {
    const int li = threadIdx.x + k * ATHREADS;
    s_lvl[li] = levels[toff + base + li];
  }
#endif
  __syncthreads();

  float rAcc[KPT];
  int   rPtr[KPT];
  for (int k = 0; k < KPT; ++k) {
    const int li = threadIdx.x + k * ATHREADS;
    const int gi = base + li;
    const float a  = attr[toff + gi];
    const float lv = s_lvl[li];
    const int   p  = parent[toff + gi];
    float z = BETA_F * (a - thv);
    z = fminf(CLAMP_V, fmaxf(-CLAMP_V, z));
    const float sig = 1.0f / (1.0f + __expf(-z));
    if (gi == 0) {                    // root: c = levels[0], pointer -> sentinel
      rAcc[k] = lv;
      rPtr[k] = N_NODES;
    } else {                          // parent[i] < i, so in-block parent is in LDS
      const float lp = (p >= base) ? s_lvl[p - base] : levels[toff + p];
      rAcc[k] = sig * (lv - lp);
      rPtr[k] = p;
    }
    Pair pr; pr.acc = rAcc[k]; pr.ptr = rPtr[k];
    s_pair[li] = pr;
  }
  __syncthreads();

  // Synchronized doubling: after AITERS steps every pointer exits the block.
  for (int it = 0; it < AITERS; ++it) {
    Pair tg[KPT];
    for (int k = 0; k < KPT; ++k) {
      const int p = rPtr[k];
      if (p >= base && p < base + BE) {
        tg[k] = s_pair[p - base];
      } else {
        tg[k].acc = 0.0f; tg[k].ptr = p;   // frozen: no change
      }
    }
    __syncthreads();
    for (int k = 0; k < KPT; ++k) {
      rAcc[k] += tg[k].acc;
      rPtr[k]  = tg[k].ptr;
      const int li = threadIdx.x + k * ATHREADS;
      Pair pr; pr.acc = rAcc[k]; pr.ptr = rPtr[k];
      s_pair[li] = pr;
    }
    __syncthreads();
  }

  Pair* outT = outPairs + (size_t)t * NP1;
  for (int k = 0; k < KPT; ++k) {
    const int li = threadIdx.x + k * ATHREADS;
    Pair pr; pr.acc = rAcc[k]; pr.ptr = rPtr[k];
    outT[base + li] = pr;
  }
  if (b == 0 && threadIdx.x == 0) {   // sentinel: acc=0, ptr=N (self-loop, identity)
    Pair s; s.acc = 0.0f; s.ptr = N_NODES;
    outT[N_NODES] = s;
  }
}

// ---------------- Phase B: global pointer-doubling on packed 8B pairs ----------------
__global__ __launch_bounds__(256) void jump_kernel(
    const unsigned long long* __restrict__ in, unsigned long long* __restrict__ out) {
  const int t = blockIdx.y;
  const int n = blockIdx.x * 256 + threadIdx.x;
  if (n >= NP1) return;
  const unsigned long long* pi = in + (size_t)t * NP1;
  const unsigned long long me = pi[n];                 // {acc:f32 lo, ptr:i32 hi} one b64
  const int mptr = (int)(me >> 32);
  const unsigned long long tg = pi[mptr];              // one b64 gather (L2-resident)
  const float acc = __uint_as_float((unsigned)me) + __uint_as_float((unsigned)tg);
  out[(size_t)t * NP1 + n] =
      (tg & 0xFFFFFFFF00000000ull) | (unsigned long long)__float_as_uint(acc);
}

// ---------------- Output: per-pixel gather, 4 pixels/thread ----------------
__global__ __launch_bounds__(256) void gather_kernel(
    const Pair* __restrict__ v, const int* __restrict__ p2n, float* __restrict__ out) {
  const int t = blockIdx.y;
  const size_t basePix = (size_t)t * HW_PIX;
  const int p4 = (blockIdx.x * 256 + threadIdx.x) * 4;
  const int4 idx = *(const int4*)(p2n + basePix + p4);
  const Pair* vt = v + (size_t)t * NP1;
  float4 r;
  r.x = vt[idx.x].acc;
  r.y = vt[idx.y].acc;
  r.z = vt[idx.z].acc;
  r.w = vt[idx.w].acc;
  *(float4*)(out + basePix + p4) = r;
}

// ---------------- launch ----------------
extern "C" void kernel_launch(void* const* d_in, const int* in_sizes, int n_in,
                              void* d_out, int out_size, void* d_ws, size_t ws_size,
                              hipStream_t stream) {
  // setup_inputs order: x(unused), attr_norm, levels, thr, parent, pixel_to_node
  const float* attr   = (const float*)d_in[1];
  const float* levels = (const float*)d_in[2];
  const float* thr    = (const float*)d_in[3];
  const int*   parent = (const int*)d_in[4];
  const int*   p2n    = (const int*)d_in[5];
  float* out = (float*)d_out;

  Pair* bufA = (Pair*)d_ws;                      // T*NP1 pairs (~8.4 MB)
  Pair* bufB = bufA + (size_t)T_TREES * NP1;     // ping-pong partner

  phaseA_kernel<<<dim3(BPT, T_TREES), ATHREADS, 0, stream>>>(attr, levels, parent, thr, bufA);

  Pair* cur = bufA;
  Pair* nxt = bufB;
  for (int i = 0; i < BITERS; ++i) {
    jump_kernel<<<dim3((NP1 + 255) / 256, T_TREES), 256, 0, stream>>>(
        (const unsigned long long*)cur, (unsigned long long*)nxt);
    Pair* tmp = cur; cur = nxt; nxt = tmp;       // BITERS even -> ends in bufA
  }

  gather_kernel<<<dim3(HW_PIX / 4 / 256, T_TREES), 256, 0, stream>>>(cur, p2n, out);
}